// GCN_20615843021630
// MI455X (gfx1250) — compile-verified
//
#include <hip/hip_runtime.h>
#include <hip/hip_bf16.h>

typedef float v2f __attribute__((ext_vector_type(2)));
typedef float v8f __attribute__((ext_vector_type(8)));

#define HIDF 128
#define LDSTRIDE 132   // pad: 132 % 64 banks = 4 -> conflict-free column reads

// ---------------------------------------------------------------------------
// Fill kernel (replaces hipMemsetAsync; graph-capture safe)
// ---------------------------------------------------------------------------
__global__ __launch_bounds__(256) void fill_f32(float* __restrict__ p, float v, int n) {
    int i = blockIdx.x * 256 + threadIdx.x;
    if (i < n) p[i] = v;
}

// ---------------------------------------------------------------------------
// Degree accumulation over dst edges (deg pre-filled with 1.0 for self loops)
// ---------------------------------------------------------------------------
__global__ __launch_bounds__(256) void deg_kernel(const int* __restrict__ dst,
                                                  float* __restrict__ deg, int e) {
    int i = blockIdx.x * 256 + threadIdx.x;
    if (i < e) atomicAdd(&deg[dst[i]], 1.0f);
}

__global__ __launch_bounds__(256) void dinv_kernel(float* __restrict__ deg, int n) {
    int i = blockIdx.x * 256 + threadIdx.x;
    if (i < n) deg[i] = __frsqrt_rn(fmaxf(deg[i], 1.0f));   // in-place -> dinv
}

// ---------------------------------------------------------------------------
// H = X @ W  via V_WMMA_F32_16X16X4_F32.
// Block = 256 threads = 8 waves. Block computes a 16x128 row-stripe of H.
// X tile (16x128 f32) staged in LDS once; each wave owns one 16-col tile of W.
// ---------------------------------------------------------------------------
__global__ __launch_bounds__(256) void gemm128_wmma(const float* __restrict__ X,
                                                    const float* __restrict__ W,
                                                    float* __restrict__ H, int nrows) {
    __shared__ float As[16 * LDSTRIDE];

    const int row0 = blockIdx.x * 16;

    // cooperative stage of the 16x128 X tile into LDS
    for (int i = threadIdx.x; i < 16 * HIDF; i += 256) {
        int r = i >> 7, c = i & (HIDF - 1);
        int gr = row0 + r;
        As[r * LDSTRIDE + c] = (gr < nrows) ? X[(size_t)gr * HIDF + c] : 0.0f;
    }
    __syncthreads();

    const int wave  = threadIdx.x >> 5;          // 0..7 -> output col tile
    const int lane  = threadIdx.x & 31;
    const int col0  = wave * 16;
    const int m     = lane & 15;                 // A row / B col within tile
    const int khalf = (lane >> 4) * 2;           // lanes 16..31 carry K+2..K+3

    v8f c = {};
    #pragma unroll
    for (int k = 0; k < HIDF; k += 4) {
        v2f a, b;
        // A 16x4: VGPR0 = K=khalf, VGPR1 = K=khalf+1  (per-lane row m)
        a.x = As[m * LDSTRIDE + k + khalf + 0];
        a.y = As[m * LDSTRIDE + k + khalf + 1];
        // B 4x16: lane holds column col0+m, rows K=khalf, khalf+1 (W row-major KxN)
        b.x = W[(size_t)(k + khalf + 0) * HIDF + col0 + m];
        b.y = W[(size_t)(k + khalf + 1) * HIDF + col0 + m];
        c = __builtin_amdgcn_wmma_f32_16x16x4_f32(
                /*neg_a=*/false, a, /*neg_b=*/false, b,
                /*c_mod=*/(short)0, c, /*reuse_a=*/false, /*reuse_b=*/false);
    }

    // C/D layout: VGPR r -> row r + 8*(lane>=16), col = lane&15
    const int col = col0 + m;
    #pragma unroll
    for (int r = 0; r < 8; ++r) {
        int gr = row0 + r + ((lane >> 4) << 3);
        if (gr < nrows) H[(size_t)gr * HIDF + col] = c[r];
    }
}

// ---------------------------------------------------------------------------
// Edge scatter: agg[dst] += dinv[src]*dinv[dst] * t[src]
// One wave32 per edge; lane moves float4 -> 512B/row fully coalesced.
// ---------------------------------------------------------------------------
__global__ __launch_bounds__(256) void scatter_edges(const int* __restrict__ src,
                                                     const int* __restrict__ dst,
                                                     const float* __restrict__ dinv,
                                                     const float* __restrict__ t,
                                                     float* __restrict__ agg, int e) {
    int wid  = (blockIdx.x * 256 + threadIdx.x) >> 5;
    int lane = threadIdx.x & 31;
    if (wid >= e) return;
    int s = src[wid], d = dst[wid];
    float w = dinv[s] * dinv[d];
    const float4* hp = (const float4*)(t + (size_t)s * HIDF);
    float4 v = hp[lane];
    float* ap = agg + (size_t)d * HIDF + lane * 4;
    atomicAdd(ap + 0, w * v.x);
    atomicAdd(ap + 1, w * v.y);
    atomicAdd(ap + 2, w * v.z);
    atomicAdd(ap + 3, w * v.w);
}

// ---------------------------------------------------------------------------
// agg = agg + dinv^2 * t + b ; optional PReLU
// ---------------------------------------------------------------------------
__global__ __launch_bounds__(256) void finalize_kernel(float* __restrict__ agg,
                                                       const float* __restrict__ t,
                                                       const float* __restrict__ dinv,
                                                       const float* __restrict__ b,
                                                       const float* __restrict__ a,
                                                       int nelem, int do_prelu) {
    int i = blockIdx.x * 256 + threadIdx.x;
    if (i >= nelem) return;
    int node = i >> 7, f = i & (HIDF - 1);
    float di = dinv[node];
    float v = agg[i] + di * di * t[i] + b[f];
    if (do_prelu) v = (v > 0.0f) ? v : a[0] * v;
    agg[i] = v;
}

// ---------------------------------------------------------------------------
// Mean-pool accumulation: pooled[batch[n]] += h[n]; cnt[batch[n]] += 1
// One wave32 per node (float4 per lane).
// ---------------------------------------------------------------------------
__global__ __launch_bounds__(256) void pool_kernel(const float* __restrict__ h,
                                                   const int* __restrict__ batch,
                                                   float* __restrict__ pooled,
                                                   float* __restrict__ cnt, int n) {
    int wid  = (blockIdx.x * 256 + threadIdx.x) >> 5;
    int lane = threadIdx.x & 31;
    if (wid >= n) return;
    int g = batch[wid];
    const float4* hp = (const float4*)(h + (size_t)wid * HIDF);
    float4 v = hp[lane];
    float* pp = pooled + (size_t)g * HIDF + lane * 4;
    atomicAdd(pp + 0, v.x);
    atomicAdd(pp + 1, v.y);
    atomicAdd(pp + 2, v.z);
    atomicAdd(pp + 3, v.w);
    if (lane == 0) atomicAdd(&cnt[g], 1.0f);
}

// ---------------------------------------------------------------------------
// out[g,o] = (pooled[g,:]/cnt[g]) @ linW[:,o] + linb[o]   (64x10, tiny)
// ---------------------------------------------------------------------------
__global__ __launch_bounds__(64) void head_kernel(const float* __restrict__ pooled,
                                                  const float* __restrict__ cnt,
                                                  const float* __restrict__ linW,
                                                  const float* __restrict__ linb,
                                                  float* __restrict__ out,
                                                  int ngraphs, int odim) {
    int i = blockIdx.x * 64 + threadIdx.x;
    if (i >= ngraphs * odim) return;
    int g = i / odim, o = i % odim;
    float inv = 1.0f / fmaxf(cnt[g], 1.0f);
    float acc = linb[o];
    #pragma unroll 4
    for (int k = 0; k < HIDF; ++k)
        acc = fmaf(pooled[g * HIDF + k] * inv, linW[k * odim + o], acc);
    out[i] = acc;
}

// ---------------------------------------------------------------------------
extern "C" void kernel_launch(void* const* d_in, const int* in_sizes, int n_in,
                              void* d_out, int out_size, void* d_ws, size_t ws_size,
                              hipStream_t stream) {
    const float* x    = (const float*)d_in[0];
    const int*   ei   = (const int*)d_in[1];
    const int*   bat  = (const int*)d_in[2];
    const float* W0   = (const float*)d_in[3];
    const float* b0   = (const float*)d_in[4];
    const float* a0   = (const float*)d_in[5];
    const float* W1   = (const float*)d_in[6];
    const float* b1   = (const float*)d_in[7];
    const float* a1   = (const float*)d_in[8];
    const float* W2   = (const float*)d_in[9];
    const float* b2   = (const float*)d_in[10];
    const float* linW = (const float*)d_in[11];
    const float* linb = (const float*)d_in[12];

    const int n = in_sizes[0] / HIDF;     // nodes
    const int e = in_sizes[1] / 2;        // edges
    const int odim    = in_sizes[12];     // 10
    const int ngraphs = out_size / odim;  // 64

    const int* src = ei;
    const int* dst = ei + e;

    // workspace carve-up
    char*  ws = (char*)d_ws;
    size_t off = 0;
    auto carve = [&](size_t bytes) {
        void* p = ws + off;
        off += (bytes + 255) & ~(size_t)255;
        return p;
    };
    float* buf0   = (float*)carve((size_t)n * HIDF * 4);   // t = h @ W
    float* buf1   = (float*)carve((size_t)n * HIDF * 4);   // agg / activations
    float* dinv   = (float*)carve((size_t)n * 4);
    float* pooled = (float*)carve((size_t)ngraphs * HIDF * 4);
    float* cnt    = (float*)carve((size_t)ngraphs * 4);
    (void)ws_size;

    const int nelem      = n * HIDF;
    const int gFillNode  = (n + 255) / 256;
    const int gFillElem  = (nelem + 255) / 256;
    const int gEdgeTh    = (e + 255) / 256;
    const int gRowTiles  = (n + 15) / 16;
    const int gEdgeWave  = (e * 32 + 255) / 256;   // one wave per edge
    const int gNodeWave  = (n * 32 + 255) / 256;   // one wave per node

    // --- symmetric normalization ---
    fill_f32<<<gFillNode, 256, 0, stream>>>(dinv, 1.0f, n);          // self-loop degree
    deg_kernel<<<gEdgeTh, 256, 0, stream>>>(dst, dinv, e);
    dinv_kernel<<<gFillNode, 256, 0, stream>>>(dinv, n);

    // --- layer 0: t = x@W0 ; agg = scatter ; h1 = prelu(agg + dinv^2 t + b0) ---
    gemm128_wmma<<<gRowTiles, 256, 0, stream>>>(x, W0, buf0, n);
    fill_f32<<<gFillElem, 256, 0, stream>>>(buf1, 0.0f, nelem);
    scatter_edges<<<gEdgeWave, 256, 0, stream>>>(src, dst, dinv, buf0, buf1, e);
    finalize_kernel<<<gFillElem, 256, 0, stream>>>(buf1, buf0, dinv, b0, a0, nelem, 1);

    // --- layer 1 ---
    gemm128_wmma<<<gRowTiles, 256, 0, stream>>>(buf1, W1, buf0, n);
    fill_f32<<<gFillElem, 256, 0, stream>>>(buf1, 0.0f, nelem);
    scatter_edges<<<gEdgeWave, 256, 0, stream>>>(src, dst, dinv, buf0, buf1, e);
    finalize_kernel<<<gFillElem, 256, 0, stream>>>(buf1, buf0, dinv, b1, a1, nelem, 1);

    // --- layer 2 (no activation) ---
    gemm128_wmma<<<gRowTiles, 256, 0, stream>>>(buf1, W2, buf0, n);
    fill_f32<<<gFillElem, 256, 0, stream>>>(buf1, 0.0f, nelem);
    scatter_edges<<<gEdgeWave, 256, 0, stream>>>(src, dst, dinv, buf0, buf1, e);
    finalize_kernel<<<gFillElem, 256, 0, stream>>>(buf1, buf0, dinv, b2, a0, nelem, 0);

    // --- global mean pool + linear head ---
    fill_f32<<<(ngraphs * HIDF + 255) / 256, 256, 0, stream>>>(pooled, 0.0f, ngraphs * HIDF);
    fill_f32<<<1, 256, 0, stream>>>(cnt, 0.0f, ngraphs);
    pool_kernel<<<gNodeWave, 256, 0, stream>>>(buf1, bat, pooled, cnt, n);
    head_kernel<<<(ngraphs * odim + 63) / 64, 64, 0, stream>>>(
        pooled, cnt, linW, linb, (float*)d_out, ngraphs, odim);
}